// SymmetricContraction_19172734009574
// MI455X (gfx1250) — compile-verified
//
#include <hip/hip_runtime.h>

// CDNA5 fp32 WMMA fragment types (wave32)
typedef float v2f __attribute__((ext_vector_type(2)));
typedef float v4f __attribute__((ext_vector_type(4)));
typedef float v8f __attribute__((ext_vector_type(8)));

static __device__ __forceinline__ v2f mk2(float a, float b) {
  v2f r; r.x = a; r.y = b; return r;
}

// D = A(16x4 f32) * B(4x16 f32) + C(16x16 f32)
static __device__ __forceinline__ v8f wmma_f32_k4(v2f a, v2f b, v8f c) {
  return __builtin_amdgcn_wmma_f32_16x16x4_f32(
      /*neg_a=*/false, a, /*neg_b=*/false, b,
      /*c_mod=*/(short)0, c, /*reuse_a=*/false, /*reuse_b=*/false);
}

// ---------------------------------------------------------------------------
// Repack kernel: rewrite the ragged CG bases into exact WMMA B-fragment order.
// rep3: [ab(256)][tile(4)][lane(32)][t(4)][v(2)]   (262144 f32)
// rep2: [a(16)][tile(2)][lane(32)][t(4)][v(2)]     (  8192 f32)
// rep1: [lane(32)][t(4)][v(2)]                     (    256 f32)
// Fragment element: row K(=d) = t*4 + v + 2*(lane>=16), col = tile*16 + lane%16
// Column blocks (k outer, i inner per irrep j): order3 widths 4*irj -> offsets
// {0,4,16,36}; order2 widths 2*irj -> {0,2,8,18}; order1 widths irj -> {0,1,4,9}.
// ---------------------------------------------------------------------------
__global__ __launch_bounds__(256) void repack_bases_kernel(
    const float* __restrict__ b1_0, const float* __restrict__ b1_1,
    const float* __restrict__ b1_2, const float* __restrict__ b1_3,
    const float* __restrict__ b2_0, const float* __restrict__ b2_1,
    const float* __restrict__ b2_2, const float* __restrict__ b2_3,
    const float* __restrict__ b3_0, const float* __restrict__ b3_1,
    const float* __restrict__ b3_2, const float* __restrict__ b3_3,
    float* __restrict__ rep3, float* __restrict__ rep2, float* __restrict__ rep1)
{
  const int gid = blockIdx.x * 256 + threadIdx.x;
  if (gid < 262144) {                       // ---- order 3 ----
    const int v    = gid & 1;
    const int t    = (gid >> 1) & 3;
    const int lane = (gid >> 3) & 31;
    const int tile = (gid >> 8) & 3;
    const int ab   = gid >> 10;
    const int a = ab >> 4, b = ab & 15;
    const int d   = t * 4 + v + ((lane >> 4) << 1);
    const int col = tile * 16 + (lane & 15);
    const int abd = ((a * 16 + b) * 16 + d) * 4;     // *4 = mul3
    float val;
    if (col < 4)       { int w = col;      val = b3_0[(abd + w) * 1]; }
    else if (col < 16) { int w = col - 4;  val = b3_1[(abd + w / 3) * 3 + w % 3]; }
    else if (col < 36) { int w = col - 16; val = b3_2[(abd + w / 5) * 5 + w % 5]; }
    else               { int w = col - 36; val = b3_3[(abd + w / 7) * 7 + w % 7]; }
    rep3[gid] = val;
  } else if (gid < 262144 + 8192) {         // ---- order 2 ----
    const int g    = gid - 262144;
    const int v    = g & 1;
    const int t    = (g >> 1) & 3;
    const int lane = (g >> 3) & 31;
    const int tile = (g >> 8) & 1;
    const int a    = g >> 9;
    const int d   = t * 4 + v + ((lane >> 4) << 1);
    const int col = tile * 16 + (lane & 15);          // 0..31
    const int ad  = (a * 16 + d) * 2;                 // *2 = mul2
    float val;
    if (col < 2)       { int w = col;      val = b2_0[(ad + w) * 1]; }
    else if (col < 8)  { int w = col - 2;  val = b2_1[(ad + w / 3) * 3 + w % 3]; }
    else if (col < 18) { int w = col - 8;  val = b2_2[(ad + w / 5) * 5 + w % 5]; }
    else               { int w = col - 18; val = b2_3[(ad + w / 7) * 7 + w % 7]; }
    rep2[g] = val;
  } else if (gid < 262144 + 8192 + 256) {   // ---- order 1 ----
    const int g    = gid - (262144 + 8192);
    const int v    = g & 1;
    const int t    = (g >> 1) & 3;
    const int lane = g >> 3;
    const int d   = t * 4 + v + ((lane >> 4) << 1);
    const int col = lane & 15;
    float val;
    if (col < 1)      { val = b1_0[d]; }
    else if (col < 4) { int i = col - 1; val = b1_1[d * 3 + i]; }
    else if (col < 9) { int i = col - 4; val = b1_2[d * 5 + i]; }
    else              { int i = col - 9; val = b1_3[d * 7 + i]; }
    rep1[g] = val;
  }
}

// ---------------------------------------------------------------------------
// Main kernel: one block (128 threads / 4 waves) per node.
// Each wave register-blocks M=32 channels (two 16-row A tiles) so every
// B-fragment load is reused by 2x the WMMA ops -> halves L2 B-traffic.
// Per-channel monomials x[c,a]*x[c,b] are folded into the A operand
// (A rows == channels), making every order a pure chained-WMMA GEMM.
// ---------------------------------------------------------------------------
__global__ __launch_bounds__(128) void symcon_kernel(
    const float* __restrict__ x, const int* __restrict__ species,
    const float* __restrict__ w1_0, const float* __restrict__ w1_1,
    const float* __restrict__ w1_2, const float* __restrict__ w1_3,
    const float* __restrict__ w2_0, const float* __restrict__ w2_1,
    const float* __restrict__ w2_2, const float* __restrict__ w2_3,
    const float* __restrict__ w3_0, const float* __restrict__ w3_1,
    const float* __restrict__ w3_2, const float* __restrict__ w3_3,
    const float* __restrict__ rep3, const float* __restrict__ rep2,
    const float* __restrict__ rep1, float* __restrict__ out)
{
  __shared__ float lds[128 * 112];   // 57344 B; first 2048 f32 reused to stage x[n]
  const int tid  = threadIdx.x;
  const int n    = blockIdx.x;
  const int lane = tid & 31;
  const int half = lane >> 4;
  const int wv   = tid >> 5;          // wave id 0..3, 32 channels each
  const int c0   = wv * 32;
  const int cc0  = c0 + (lane & 15);  // channel of A-tile 0 row
  const int cc1  = cc0 + 16;          // channel of A-tile 1 row

  // stage x[n] (128x16 f32 = 8 KB) into LDS: 128 threads x 4 v4f
  {
    const v4f* xg = (const v4f*)(x + (size_t)n * 2048);
    v4f* xl = (v4f*)lds;
#pragma unroll
    for (int u = 0; u < 4; ++u) xl[tid * 4 + u] = xg[tid * 4 + u];
  }
  __syncthreads();

  // A fragments for both channel tiles: afr[t] = x[cc, 4t + v + 2*half]
  v2f afrA[4], afrB[4];
#pragma unroll
  for (int t = 0; t < 4; ++t) {
    afrA[t] = *(const v2f*)&lds[cc0 * 16 + t * 4 + half * 2];
    afrB[t] = *(const v2f*)&lds[cc1 * 16 + t * 4 + half * 2];
  }

  const v8f zv = {0, 0, 0, 0, 0, 0, 0, 0};
  v8f m3A[4] = {zv, zv, zv, zv}, m3B[4] = {zv, zv, zv, zv};
  v8f m2A[2] = {zv, zv},         m2B[2] = {zv, zv};
  v8f m1A = zv,                  m1B = zv;

  // ---- order 1: M1 = X · B1 ----
  {
    const float* bp = rep1 + lane * 8;
    const v4f lo = *(const v4f*)(bp);
    const v4f hi = *(const v4f*)(bp + 4);
    const v2f f0 = mk2(lo.x, lo.y), f1 = mk2(lo.z, lo.w);
    const v2f f2 = mk2(hi.x, hi.y), f3 = mk2(hi.z, hi.w);
    m1A = wmma_f32_k4(afrA[0], f0, m1A);  m1B = wmma_f32_k4(afrB[0], f0, m1B);
    m1A = wmma_f32_k4(afrA[1], f1, m1A);  m1B = wmma_f32_k4(afrB[1], f1, m1B);
    m1A = wmma_f32_k4(afrA[2], f2, m1A);  m1B = wmma_f32_k4(afrB[2], f2, m1B);
    m1A = wmma_f32_k4(afrA[3], f3, m1A);  m1B = wmma_f32_k4(afrB[3], f3, m1B);
  }

  // ---- order 2: M2 = Σ_a (diag(x_:,a)·X) · B2[a] ----
  for (int a = 0; a < 16; ++a) {
    const float paA = lds[cc0 * 16 + a];
    const float paB = lds[cc1 * 16 + a];
    v2f asA[4], asB[4];
#pragma unroll
    for (int t = 0; t < 4; ++t) { asA[t] = afrA[t] * paA; asB[t] = afrB[t] * paB; }
#pragma unroll
    for (int tile = 0; tile < 2; ++tile) {
      const float* bp = rep2 + (size_t)((a * 2 + tile) * 32 + lane) * 8;
      const v4f lo = *(const v4f*)(bp);
      const v4f hi = *(const v4f*)(bp + 4);
      const v2f f0 = mk2(lo.x, lo.y), f1 = mk2(lo.z, lo.w);
      const v2f f2 = mk2(hi.x, hi.y), f3 = mk2(hi.z, hi.w);
      m2A[tile] = wmma_f32_k4(asA[0], f0, m2A[tile]);
      m2B[tile] = wmma_f32_k4(asB[0], f0, m2B[tile]);
      m2A[tile] = wmma_f32_k4(asA[1], f1, m2A[tile]);
      m2B[tile] = wmma_f32_k4(asB[1], f1, m2B[tile]);
      m2A[tile] = wmma_f32_k4(asA[2], f2, m2A[tile]);
      m2B[tile] = wmma_f32_k4(asB[2], f2, m2B[tile]);
      m2A[tile] = wmma_f32_k4(asA[3], f3, m2A[tile]);
      m2B[tile] = wmma_f32_k4(asB[3], f3, m2B[tile]);
    }
  }

  // ---- order 3: M3 = Σ_{a,b} (diag(x_:,a·x_:,b)·X) · B3[a,b] ----
  for (int a = 0; a < 16; ++a) {
    const float paA = lds[cc0 * 16 + a];
    const float paB = lds[cc1 * 16 + a];
    for (int b = 0; b < 16; ++b) {
      const float pabA = paA * lds[cc0 * 16 + b];
      const float pabB = paB * lds[cc1 * 16 + b];
      v2f asA[4], asB[4];
#pragma unroll
      for (int t = 0; t < 4; ++t) { asA[t] = afrA[t] * pabA; asB[t] = afrB[t] * pabB; }
      const int ab = a * 16 + b;
#pragma unroll
      for (int tile = 0; tile < 4; ++tile) {
        const float* bp = rep3 + (size_t)((ab * 4 + tile) * 32 + lane) * 8;
        const v4f lo = *(const v4f*)(bp);
        const v4f hi = *(const v4f*)(bp + 4);
        const v2f f0 = mk2(lo.x, lo.y), f1 = mk2(lo.z, lo.w);
        const v2f f2 = mk2(hi.x, hi.y), f3 = mk2(hi.z, hi.w);
        m3A[tile] = wmma_f32_k4(asA[0], f0, m3A[tile]);
        m3B[tile] = wmma_f32_k4(asB[0], f0, m3B[tile]);
        m3A[tile] = wmma_f32_k4(asA[1], f1, m3A[tile]);
        m3B[tile] = wmma_f32_k4(asB[1], f1, m3B[tile]);
        m3A[tile] = wmma_f32_k4(asA[2], f2, m3A[tile]);
        m3B[tile] = wmma_f32_k4(asB[2], f2, m3B[tile]);
        m3A[tile] = wmma_f32_k4(asA[3], f3, m3A[tile]);
        m3B[tile] = wmma_f32_k4(asB[3], f3, m3B[tile]);
      }
    }
  }

  __syncthreads();   // x staging region is dead; LDS becomes moment buffer

  // scatter D-layout accumulators to Mbuf[c][112]: [0,64)=M3 [64,96)=M2 [96,112)=M1
  {
    const int mcol = lane & 15;
#pragma unroll
    for (int r = 0; r < 8; ++r) {
      const int crow0 = c0 + r + half * 8;           // D-layout row -> channel
      float* mbA = &lds[crow0 * 112];
      float* mbB = &lds[(crow0 + 16) * 112];
#pragma unroll
      for (int tile = 0; tile < 4; ++tile) {
        mbA[tile * 16 + mcol] = m3A[tile][r];
        mbB[tile * 16 + mcol] = m3B[tile][r];
      }
      mbA[64 + mcol] = m2A[0][r];  mbB[64 + mcol] = m2B[0][r];
      mbA[80 + mcol] = m2A[1][r];  mbB[80 + mcol] = m2B[1][r];
      mbA[96 + mcol] = m1A[r];     mbB[96 + mcol] = m1B[r];
    }
  }
  __syncthreads();

  // per-species weight combine:  out[c,i] = Σ_o norm_o Σ_k W_o[s,k,c]·M_o[c,(k,i)]
  const int s = species[n];
  const float norm3 = 0.5f;                    // 4^-0.5
  const float norm2 = 0.70710678118654752f;    // 2^-0.5
#pragma unroll
  for (int q = 0; q < 16; ++q) {
    const int o   = tid * 16 + q;              // 0..2047 over (c, col)
    const int c   = o >> 4;
    const int col = o & 15;
    int i, irj, o3, o2, o1;
    const float *wp3, *wp2, *wp1;
    if (col < 1)      { i = col;     irj = 1; o3 = 0;  o2 = 0;  o1 = 0; wp3 = w3_0; wp2 = w2_0; wp1 = w1_0; }
    else if (col < 4) { i = col - 1; irj = 3; o3 = 4;  o2 = 2;  o1 = 1; wp3 = w3_1; wp2 = w2_1; wp1 = w1_1; }
    else if (col < 9) { i = col - 4; irj = 5; o3 = 16; o2 = 8;  o1 = 4; wp3 = w3_2; wp2 = w2_2; wp1 = w1_2; }
    else              { i = col - 9; irj = 7; o3 = 36; o2 = 18; o1 = 9; wp3 = w3_3; wp2 = w2_3; wp1 = w1_3; }
    const float* M = &lds[c * 112];
    float a3 = 0.f;
#pragma unroll
    for (int k = 0; k < 4; ++k) a3 += wp3[(s * 4 + k) * 128 + c] * M[o3 + k * irj + i];
    float a2 = 0.f;
#pragma unroll
    for (int k = 0; k < 2; ++k) a2 += wp2[(s * 2 + k) * 128 + c] * M[64 + o2 + k * irj + i];
    const float a1 = wp1[s * 128 + c] * M[96 + o1 + i];
    out[(size_t)n * 2048 + o] = a3 * norm3 + a2 * norm2 + a1;
  }
}

// ---------------------------------------------------------------------------
// d_in layout (pytree order of setup_inputs):
//   [0] x (512*128*16 f32)   [1] index (512 i32)
//   [2..5]  weights order1 j0..j3   (64*1*128)
//   [6..9]  weights order2 j0..j3   (64*2*128)
//   [10..13] weights order3 j0..j3  (64*4*128)
//   [14..17] bases order1 j0..j3    (16*1*irj)
//   [18..21] bases order2 j0..j3    (16*16*2*irj)
//   [22..25] bases order3 j0..j3    (16*16*16*4*irj)
// ---------------------------------------------------------------------------
extern "C" void kernel_launch(void* const* d_in, const int* in_sizes, int n_in,
                              void* d_out, int out_size, void* d_ws, size_t ws_size,
                              hipStream_t stream) {
  (void)in_sizes; (void)n_in; (void)out_size; (void)ws_size;
  const float* x  = (const float*)d_in[0];
  const int*   si = (const int*)d_in[1];
  const float* w1_0 = (const float*)d_in[2],  *w1_1 = (const float*)d_in[3];
  const float* w1_2 = (const float*)d_in[4],  *w1_3 = (const float*)d_in[5];
  const float* w2_0 = (const float*)d_in[6],  *w2_1 = (const float*)d_in[7];
  const float* w2_2 = (const float*)d_in[8],  *w2_3 = (const float*)d_in[9];
  const float* w3_0 = (const float*)d_in[10], *w3_1 = (const float*)d_in[11];
  const float* w3_2 = (const float*)d_in[12], *w3_3 = (const float*)d_in[13];
  const float* b1_0 = (const float*)d_in[14], *b1_1 = (const float*)d_in[15];
  const float* b1_2 = (const float*)d_in[16], *b1_3 = (const float*)d_in[17];
  const float* b2_0 = (const float*)d_in[18], *b2_1 = (const float*)d_in[19];
  const float* b2_2 = (const float*)d_in[20], *b2_3 = (const float*)d_in[21];
  const float* b3_0 = (const float*)d_in[22], *b3_1 = (const float*)d_in[23];
  const float* b3_2 = (const float*)d_in[24], *b3_3 = (const float*)d_in[25];

  float* rep3 = (float*)d_ws;          // 262144 f32
  float* rep2 = rep3 + 262144;         //   8192 f32
  float* rep1 = rep2 + 8192;           //    256 f32  (total ~1.06 MB of d_ws)

  const int total = 262144 + 8192 + 256;
  repack_bases_kernel<<<(total + 255) / 256, 256, 0, stream>>>(
      b1_0, b1_1, b1_2, b1_3, b2_0, b2_1, b2_2, b2_3,
      b3_0, b3_1, b3_2, b3_3, rep3, rep2, rep1);

  symcon_kernel<<<512, 128, 0, stream>>>(
      x, si,
      w1_0, w1_1, w1_2, w1_3, w2_0, w2_1, w2_2, w2_3, w3_0, w3_1, w3_2, w3_3,
      rep3, rep2, rep1, (float*)d_out);
}